// GRNGCN_78563541778798
// MI455X (gfx1250) — compile-verified
//
#include <hip/hip_runtime.h>
#include <hip/hip_bf16.h>

#define N_NODES 50000

typedef __attribute__((ext_vector_type(2))) float v2f;
typedef __attribute__((ext_vector_type(8))) float v8f;

// Guaranteed-native fp32 global atomic add (CDNA5 GLOBAL_ATOMIC_ADD_F32,
// non-returning form -> STOREcnt; S_ENDPGM implicitly waits for idle).
__device__ __forceinline__ void atom_add_f32(float* p, float v) {
  asm volatile("global_atomic_add_f32 %0, %1, off" : : "v"(p), "v"(v) : "memory");
}

// ---------------------------------------------------------------------------
// Degree / normalization: deg starts at 1 (self-loop), count incoming edges,
// then dinv = rsqrt(deg) in place.
// ---------------------------------------------------------------------------
__global__ void gcn_deg_init_kernel(float* __restrict__ deg, int n) {
  int i = blockIdx.x * blockDim.x + threadIdx.x;
  if (i < n) deg[i] = 1.0f;
}

__global__ void gcn_deg_count_kernel(const int* __restrict__ dst,
                                     float* __restrict__ deg, int nE) {
  int e = blockIdx.x * blockDim.x + threadIdx.x;
  if (e < nE) atom_add_f32(&deg[dst[e]], 1.0f);
}

__global__ void gcn_deg_rsqrt_kernel(float* __restrict__ deg, int n) {
  int i = blockIdx.x * blockDim.x + threadIdx.x;
  if (i < n) deg[i] = rsqrtf(deg[i]);
}

// ---------------------------------------------------------------------------
// Fused fp32 WMMA GEMM:  H = act(A) @ B,  Hraw = H,
//                        Hagg = H * dinv[row]^2 + bias[col]   (self-loop seed)
// One wave computes a 16x16 output tile via V_WMMA_F32_16X16X4_F32 over K.
// Block = 4 waves covering 64 consecutive columns (shared A panel -> cache).
// Requires M%16==0, N%16==0, K%4==0; grid sized so EXEC is always all-1s.
// RELUA is a template parameter so the no-relu instantiation has a clean
// load->wmma inner loop (no per-element cndmask).
// ---------------------------------------------------------------------------
template <int RELUA>
__global__ __launch_bounds__(128)
void gcn_wmma_gemm_kernel(const float* __restrict__ A,
                          const float* __restrict__ B,
                          const float* __restrict__ bias,
                          const float* __restrict__ dinv,
                          float* __restrict__ Hraw,
                          float* __restrict__ Hagg,
                          int N, int K) {
  const int lane = threadIdx.x & 31;
  const int wave = threadIdx.x >> 5;
  const int r    = lane & 15;   // M/N index within tile
  const int half = lane >> 4;   // K-half selector
  const int m0   = blockIdx.x << 4;
  const int n0   = ((blockIdx.y << 2) + wave) << 4;

  v8f acc = {};
  const float* Ap = A + (size_t)(m0 + r) * K + (half << 1);
  const float* Bp = B + (size_t)(half << 1) * N + (n0 + r);

  for (int k0 = 0; k0 < K; k0 += 4) {
    // A fragment: lane holds A[m0+r][k0+2*half .. +1] (8B-aligned float2)
    v2f a = *(const v2f*)(Ap + k0);
    if (RELUA) { a.x = fmaxf(a.x, 0.0f); a.y = fmaxf(a.y, 0.0f); }
    // B fragment: lane holds B[k0+2*half][n0+r], B[k0+2*half+1][n0+r]
    v2f b;
    b.x = Bp[(size_t)k0 * N];
    b.y = Bp[(size_t)(k0 + 1) * N];
    acc = __builtin_amdgcn_wmma_f32_16x16x4_f32(
        /*neg_a=*/false, a, /*neg_b=*/false, b,
        /*c_mod=*/(short)0, acc, /*reuse_a=*/false, /*reuse_b=*/false);
  }

  const float bcol = bias[n0 + r];
#pragma unroll
  for (int v = 0; v < 8; ++v) {
    const int row  = m0 + v + (half << 3);      // acc[v] -> C[m0+v+8*half][n0+r]
    const float dv = dinv[row];
    const float h  = acc[v];
    const size_t idx = (size_t)row * N + (n0 + r);
    Hraw[idx] = h;
    Hagg[idx] = fmaf(h, dv * dv, bcol);
  }
}

// ---------------------------------------------------------------------------
// Edge scatter: Agg[dst] += H[src] * dinv[src]*dinv[dst]
// C = 4<<CH4SHIFT channels; one thread = one float4 chunk of one edge.
// CH4SHIFT=5 (C=128): one wave per edge, fully coalesced 512B row gather.
// ---------------------------------------------------------------------------
template <int CH4SHIFT>
__global__ void gcn_scatter_kernel(const float* __restrict__ H,
                                   float* __restrict__ Agg,
                                   const int* __restrict__ src,
                                   const int* __restrict__ dst,
                                   const float* __restrict__ dinv,
                                   int nE) {
  const int t = blockIdx.x * blockDim.x + threadIdx.x;
  const int e = t >> CH4SHIFT;
  const int j = t & ((1 << CH4SHIFT) - 1);
  if (e >= nE) return;
  const int C = 4 << CH4SHIFT;
  const int s = src[e];
  const int d = dst[e];
  const float w = dinv[s] * dinv[d];
  const float4 v = *(const float4*)(H + (size_t)s * C + (j << 2));
  float* p = Agg + (size_t)d * C + (j << 2);
  atom_add_f32(p + 0, v.x * w);
  atom_add_f32(p + 1, v.y * w);
  atom_add_f32(p + 2, v.z * w);
  atom_add_f32(p + 3, v.w * w);
}

// ---------------------------------------------------------------------------
// Decode: out[e] = dot(Z[a[e]], Z[b[e]]), 64-dim fp32.
// ---------------------------------------------------------------------------
__global__ void gcn_decode_kernel(const float* __restrict__ Z,
                                  const int* __restrict__ ia,
                                  const int* __restrict__ ib,
                                  float* __restrict__ out, int n) {
  int e = blockIdx.x * blockDim.x + threadIdx.x;
  if (e >= n) return;
  const float4* za = (const float4*)(Z + (size_t)ia[e] * 64);
  const float4* zb = (const float4*)(Z + (size_t)ib[e] * 64);
  float s = 0.0f;
#pragma unroll
  for (int i = 0; i < 16; ++i) {
    float4 u = za[i];
    float4 v = zb[i];
    s = fmaf(u.x, v.x, s);
    s = fmaf(u.y, v.y, s);
    s = fmaf(u.z, v.z, s);
    s = fmaf(u.w, v.w, s);
  }
  out[e] = s;
}

// ---------------------------------------------------------------------------
// Launch
// ---------------------------------------------------------------------------
extern "C" void kernel_launch(void* const* d_in, const int* in_sizes, int n_in,
                              void* d_out, int out_size, void* d_ws, size_t ws_size,
                              hipStream_t stream) {
  const float* x  = (const float*)d_in[0];   // [50000, 256]
  const int*   ei = (const int*)d_in[1];     // [2, 600000]
  const int*   el = (const int*)d_in[2];     // [2, 200000]
  const float* W1 = (const float*)d_in[3];   // [256, 128]
  const float* b1 = (const float*)d_in[4];   // [128]
  const float* W2 = (const float*)d_in[5];   // [128, 64]
  const float* b2 = (const float*)d_in[6];   // [64]
  float* out = (float*)d_out;                // [200000]
  float* ws  = (float*)d_ws;

  const int nE = in_sizes[1] / 2;
  const int nL = in_sizes[2] / 2;
  const int* src = ei;
  const int* dst = ei + nE;

  // Workspace layout (floats):
  float* dinv = ws;                              // 50000 (deg -> dinv in place)
  float* h0   = ws + 50048;                      // 50000*128
  float* agg1 = h0 + (size_t)N_NODES * 128;      // 50000*128
  float* h2   = h0;                              // reuse h0: 50000*64
  float* agg2 = h0 + (size_t)N_NODES * 64;       // 50000*64
  // total: ~12.85M floats ~= 51.4 MB

  // Degree + normalization
  gcn_deg_init_kernel<<<(N_NODES + 255) / 256, 256, 0, stream>>>(dinv, N_NODES);
  gcn_deg_count_kernel<<<(nE + 255) / 256, 256, 0, stream>>>(dst, dinv, nE);
  gcn_deg_rsqrt_kernel<<<(N_NODES + 255) / 256, 256, 0, stream>>>(dinv, N_NODES);

  // Layer 1: h0 = x@W1 ; agg1 = h0*dinv^2 + b1 ; scatter edges
  gcn_wmma_gemm_kernel<0><<<dim3(N_NODES / 16, 2), 128, 0, stream>>>(
      x, W1, b1, dinv, h0, agg1, 128, 256);
  gcn_scatter_kernel<5><<<(nE * 32 + 255) / 256, 256, 0, stream>>>(
      h0, agg1, src, dst, dinv, nE);

  // Layer 2: h2 = relu(agg1)@W2 ; agg2 = h2*dinv^2 + b2 ; scatter edges
  gcn_wmma_gemm_kernel<1><<<dim3(N_NODES / 16, 1), 128, 0, stream>>>(
      agg1, W2, b2, dinv, h2, agg2, 64, 128);
  gcn_scatter_kernel<4><<<(nE * 16 + 255) / 256, 256, 0, stream>>>(
      h2, agg2, src, dst, dinv, nE);

  // Decode
  gcn_decode_kernel<<<(nL + 255) / 256, 256, 0, stream>>>(agg2, el, el + nL, out, nL);
}